// Net_4355096838201
// MI455X (gfx1250) — compile-verified
//
#include <hip/hip_runtime.h>
#include <hip/hip_bf16.h>

// Problem constants (B=16, T=512, IDIM=80, HDIM=512)
#define IDIM   80
#define HDIM   512
#define NTOK   8192            // B*T
#define PAD    79              // IDIM-1
#define NSH    159             // 2*IDIM-1 shifts
#define XPLEN  238             // IDIM + 2*PAD

typedef __attribute__((ext_vector_type(2))) float v2f;
typedef __attribute__((ext_vector_type(8))) float v8f;

// ---------------------------------------------------------------------------
// Kernel 1: per-token cosine cross-correlation argmax + softmax gating.
// One wave32 per token, 8 waves per 256-thread block. Each wave uses its own
// LDS slice, so only program-order LDS consistency (in-order DS pipe) is
// needed; a block barrier is inserted anyway between fill and compute.
// ---------------------------------------------------------------------------
__global__ __launch_bounds__(256) void corr_attn_kernel(
    const float* __restrict__ x, const float* __restrict__ y,
    float* __restrict__ xattn)
{
    __shared__ float s_xpad[8][XPLEN];
    __shared__ float s_y[8][IDIM];

    const int lane = threadIdx.x & 31;
    const int wave = threadIdx.x >> 5;
    const int tok  = blockIdx.x * 8 + wave;      // 0..NTOK-1 (grid exact)

    float* xp = s_xpad[wave];
    float* yy = s_y[wave];
    const float* xr = x + tok * IDIM;
    const float* yr = y + tok * IDIM;

    // Fill LDS: zero pads (no overlap with payload), then x payload and y.
    for (int i = lane; i < PAD; i += 32) {
        xp[i] = 0.0f;
        xp[PAD + IDIM + i] = 0.0f;
    }
    for (int i = lane; i < IDIM; i += 32) {
        xp[PAD + i] = xr[i];
        yy[i] = yr[i];
    }
    __syncthreads();

    // ||y||^2 (every lane computes the same value via LDS broadcast reads)
    float yn2 = 0.0f;
    #pragma unroll 8
    for (int i = 0; i < IDIM; ++i) yn2 = fmaf(yy[i], yy[i], yn2);

    // Each lane owns 5 contiguous shifts: s = lane*5 + j  (159 = 32*5 - 1)
    float best_sim = -3.402823466e38f;
    int   best_s   = NSH;                 // sentinel loses all ties
    for (int j = 0; j < 5; ++j) {
        const int s = lane * 5 + j;
        if (s < NSH) {
            float dot = 0.0f, n2 = 0.0f;
            #pragma unroll 8
            for (int i = 0; i < IDIM; ++i) {
                const float xv = xp[s + i];           // stride 5 across lanes
                dot = fmaf(yy[i], xv, dot);
                n2  = fmaf(xv, xv, n2);
            }
            const float den2 = n2 * yn2;
            const float sim  = (den2 > 0.0f) ? dot * rsqrtf(den2) : 0.0f;
            if (sim > best_sim) { best_sim = sim; best_s = s; }  // first-max kept
        }
    }
    // Wave-wide argmax, tie-break to lowest shift (matches jnp.argmax)
    #pragma unroll
    for (int off = 16; off >= 1; off >>= 1) {
        const float osim = __shfl_xor(best_sim, off, 32);
        const int   os   = __shfl_xor(best_s, off, 32);
        if (osim > best_sim || (osim == best_sim && os < best_s)) {
            best_sim = osim; best_s = os;
        }
    }

    // Softmax over the 80-dim: lane owns i = lane, lane+32, lane+64
    float xa[3], sc[3];
    float lmax = -3.402823466e38f;
    #pragma unroll
    for (int r = 0; r < 3; ++r) {
        const int i = lane + 32 * r;
        if (i < IDIM) {
            const float v = xp[best_s + i];           // x_aug[i]
            xa[r] = v;
            const float s = (v * yy[i]) / 10.0f;      // score / TEMPER
            sc[r] = s;
            lmax = fmaxf(lmax, s);
        }
    }
    #pragma unroll
    for (int off = 16; off >= 1; off >>= 1)
        lmax = fmaxf(lmax, __shfl_xor(lmax, off, 32));

    float lsum = 0.0f;
    #pragma unroll
    for (int r = 0; r < 3; ++r) {
        const int i = lane + 32 * r;
        if (i < IDIM) {
            const float e = expf(sc[r] - lmax);
            sc[r] = e;
            lsum += e;
        }
    }
    #pragma unroll
    for (int off = 16; off >= 1; off >>= 1)
        lsum += __shfl_xor(lsum, off, 32);

    const float inv = 1.0f / lsum;                    // lsum >= exp(0) > 0
    float* orow = xattn + tok * IDIM;
    #pragma unroll
    for (int r = 0; r < 3; ++r) {
        const int i = lane + 32 * r;
        if (i < IDIM) orow[i] = xa[r] * sc[r] * inv;  // x_attn
    }
}

// ---------------------------------------------------------------------------
// Kernel 2: h[tok, hd] = x_attn[tok, :] . W[hd, :] + b[hd]
// Full-precision f32 WMMA (V_WMMA_F32_16X16X4_F32), 16x16 tile per wave,
// K=80 -> 20 WMMA steps. Layouts per CDNA5 ISA 7.12.2:
//   A 16x4 : lanes 0-15 hold M=lane, K={0,1}; lanes 16-31 hold M=lane-16, K={2,3}
//   B 4x16 : lanes 0-15 hold N=lane, K={0,1}; lanes 16-31 hold N=lane-16, K={2,3}
//   C 16x16: VGPR r -> rows m0+r (lanes 0-15) / m0+r+8 (lanes 16-31), col = lane&15
// EXEC is all-ones (no divergence), as WMMA requires.
// ---------------------------------------------------------------------------
__global__ __launch_bounds__(256) void wmma_gemm_kernel(
    const float* __restrict__ xattn, const float* __restrict__ W,
    const float* __restrict__ bias, float* __restrict__ out)
{
    const int lane = threadIdx.x & 31;
    const int wave = threadIdx.x >> 5;                 // 0..7
    const int n0   = blockIdx.x * 16;                  // HDIM tile
    const int m0   = (blockIdx.y * 8 + wave) * 16;     // token tile
    const int half = lane >> 4;                        // 0 or 1 (K-pair select)
    const int l    = lane & 15;

    const float* arow = xattn + (size_t)(m0 + l) * IDIM + 2 * half;
    const float* brow = W     + (size_t)(n0 + l) * IDIM + 2 * half;

    v8f c = {};
    #pragma unroll
    for (int k = 0; k < IDIM; k += 4) {
        const v2f a = *(const v2f*)(arow + k);         // A[M=l][k+2h .. k+2h+1]
        const v2f b = *(const v2f*)(brow + k);         // B[k+2h .. ][N=l] = W[n][k]
        c = __builtin_amdgcn_wmma_f32_16x16x4_f32(
                /*neg_a=*/false, a, /*neg_b=*/false, b,
                /*c_mod=*/(short)0, c, /*reuse_a=*/false, /*reuse_b=*/false);
    }

    const float bv = bias[n0 + l];
    const int mbase = m0 + half * 8;
    #pragma unroll
    for (int r = 0; r < 8; ++r)
        out[(size_t)(mbase + r) * HDIM + n0 + l] = c[r] + bv;
}

// ---------------------------------------------------------------------------
extern "C" void kernel_launch(void* const* d_in, const int* in_sizes, int n_in,
                              void* d_out, int out_size, void* d_ws, size_t ws_size,
                              hipStream_t stream)
{
    (void)in_sizes; (void)n_in; (void)out_size; (void)ws_size;
    const float* x  = (const float*)d_in[0];   // (16,512,80)
    const float* y  = (const float*)d_in[1];   // (16,512,80)
    const float* w  = (const float*)d_in[2];   // (512,80)
    const float* b  = (const float*)d_in[3];   // (512,)
    float* out   = (float*)d_out;              // (16,512,512)
    float* xattn = (float*)d_ws;               // NTOK*IDIM f32 = 2.62 MB scratch

    // Phase 1: one wave per token, 8 waves/block -> 1024 blocks
    corr_attn_kernel<<<dim3(NTOK / 8), dim3(256), 0, stream>>>(x, y, xattn);

    // Phase 2: 16x16 tile per wave; 8 waves/block cover 128 tokens.
    // grid = (HDIM/16, NTOK/128) = (32, 64)
    wmma_gemm_kernel<<<dim3(HDIM / 16, NTOK / 128), dim3(256), 0, stream>>>(
        xattn, w, b, out);
}